// ConvTree_73349451481440
// MI455X (gfx1250) — compile-verified
//
#include <hip/hip_runtime.h>
#include <hip/hip_bf16.h>

typedef __attribute__((ext_vector_type(2))) float v2f;
typedef __attribute__((ext_vector_type(4))) float v4f;
typedef __attribute__((ext_vector_type(8))) float v8f;

#define BATCH   256
#define C_IN    128
#define NPTS    1024
#define MPTS    1023
#define C_OUT   64
#define OPRIME  192          // 3 * 64 stacked output rows (k-major)
#define JROW    416          // dwords per k-pair block in packed W': 192*2 + 32 pad
                             // (416 mod 64 == 32 -> successive k-pairs land 32 banks apart)
#define WSW_DW  (64 * JROW)  // 26,624 dwords = 104 KB packed W'
#define YSTRIDE 192          // floats per n-record of Y

// ---------------------------------------------------------------------------
// Stage 0: repack W (64,128,3) -> paired layout ws_w[j][o'][2] with o'=k*64+o,
// pair j covers channels c=2j,2j+1.  This is exactly the image stage 1 copies
// verbatim into LDS.
// ---------------------------------------------------------------------------
__global__ __launch_bounds__(256) void wprep_kernel(const float* __restrict__ W,
                                                    float* __restrict__ wsw) {
    int e = blockIdx.x * 256 + threadIdx.x;       // 0 .. 24575
    if (e >= 64 * OPRIME * 2) return;
    int j  = e / 384;                             // k-pair index (c/2)
    int r  = e - j * 384;
    int op = r >> 1;                              // o' = k*64 + o
    int s  = r & 1;                               // which channel of the pair
    int c  = 2 * j + s;
    int o  = op & 63;
    int k  = op >> 6;
    wsw[j * JROW + op * 2 + s] = W[o * (C_IN * 3) + c * 3 + k];
}

// ---------------------------------------------------------------------------
// Stage 1: per batch b:  Y[b, o', n] = sum_c W'[o', c] * trees[b, c, n]
// GEMM  [192 x 128] @ [128 x 1024]  with V_WMMA_F32_16X16X4_F32.
// Grid: (8 n-tiles of 128, 256 batches), block = 256 threads = 8 waves.
// Each wave: one 16-wide n-tile, all 12 o'-tiles, K-loop of 32 x (K=4).
// Y stored n-major (192 floats / n) so stage 2 gathers contiguous records.
// ---------------------------------------------------------------------------
__global__ __launch_bounds__(256) void gemm_kernel(const float* __restrict__ trees,
                                                   const float* __restrict__ wsw,
                                                   float* __restrict__ Y) {
    __shared__ float lw[32 * JROW];               // half of W' (K-split): 52 KB

    const int b    = blockIdx.y;
    const int wave = threadIdx.x >> 5;
    const int lane = threadIdx.x & 31;
    const int lo   = lane & 15;                   // row/col within tile
    const int hi   = lane >> 4;                   // selects K pair {0,1} vs {2,3}
    const int n    = (blockIdx.x * 8 + wave) * 16 + lo;

    const float* tb = trees + (size_t)b * (C_IN * NPTS) + n;

    v8f acc[12] = {};

    for (int kh = 0; kh < 2; ++kh) {              // two K halves of 64 channels
        __syncthreads();                          // protect LDS reuse
        {   // cooperative copy of this K-half of packed W' into LDS
            const v4f* src = (const v4f*)(wsw + (size_t)kh * (32 * JROW));
            v4f*       dst = (v4f*)lw;
            for (int i = threadIdx.x; i < (32 * JROW) / 4; i += 256)
                dst[i] = src[i];
        }
        __syncthreads();

        for (int ksl = 0; ksl < 16; ++ksl) {      // 16 K-steps of 4 per half
            const int kb = (kh * 16 + ksl) * 4 + 2 * hi;   // channel base (even)

            // B fragment (4x16 f32): trees[b, kb..kb+1, n]; lanes 0-15 carry
            // K=0,1, lanes 16-31 carry K=2,3.  Two half-wave-coalesced b32 loads.
            v2f bf;
            bf.x = tb[(size_t)kb * NPTS];
            bf.y = tb[(size_t)(kb + 1) * NPTS];

            const float* lwj = lw + (ksl * 2 + hi) * JROW;  // this lane's k-pair

            #pragma unroll
            for (int mt = 0; mt < 12; ++mt) {     // 12 o'-tiles of 16
                const int row = mt * 16 + lo;
                // A fragment (16x4 f32): ds_load_b64 of the (c, c+1) pair.
                v2f a = *(const v2f*)(lwj + row * 2);
                acc[mt] = __builtin_amdgcn_wmma_f32_16x16x4_f32(
                    /*neg_a=*/false, a, /*neg_b=*/false, bf,
                    /*c_mod=*/(short)0, acc[mt],
                    /*reuse_a=*/false, /*reuse_b=*/false);
            }
        }
    }

    // D layout: lane holds column n = lo, rows M = 8*hi + v (v = VGPR 0..7).
    // Store as two float4 runs along o' -> 32B chunks into n-major Y records.
    float* yrow = Y + ((size_t)b * NPTS + n) * YSTRIDE + 8 * hi;
    #pragma unroll
    for (int mt = 0; mt < 12; ++mt) {
        float* p = yrow + mt * 16;
        const v4f* av = (const v4f*)&acc[mt];
        *(v4f*)(p)     = av[0];
        *(v4f*)(p + 4) = av[1];
    }
}

// ---------------------------------------------------------------------------
// Stage 2: out[b, o, col] = (col==0) ? 0
//        : Y[b, idx0, o] + Y[b, idx1, 64+o] + Y[b, idx2, 128+o] + bias[o]
// Gathers are contiguous 256B record slices (coalesced across the 64 o-threads
// of a column pair of waves); results transposed through LDS so the final
// global stores are 128B-contiguous along the m axis.
// Grid: (32 column-tiles of 32, 256 batches), block = 256.
// ---------------------------------------------------------------------------
__global__ __launch_bounds__(256) void gather_kernel(const float* __restrict__ Y,
                                                     const long long* __restrict__ idx,
                                                     const float* __restrict__ bias,
                                                     float* __restrict__ out) {
    __shared__ float tile[C_OUT * 33];            // 33 pad: conflict-free transpose

    const int b  = blockIdx.y;
    const int t  = threadIdx.x;
    const int o  = t & 63;                        // output channel
    const int cg = t >> 6;                        // column subgroup 0..3

    const float      bv = bias[o];
    const float*     Yb = Y + (size_t)b * NPTS * YSTRIDE;
    const long long* ib = idx + (size_t)b * (3 * MPTS);

    for (int it = 0; it < 8; ++it) {
        const int colLocal = it * 4 + cg;
        const int col = blockIdx.x * 32 + colLocal;   // 0..1023 final column
        float v = 0.0f;
        if (col != 0) {
            const int m = col - 1;
            const long long i0 = ib[3 * m + 0];
            const long long i1 = ib[3 * m + 1];
            const long long i2 = ib[3 * m + 2];
            v = Yb[(size_t)i0 * YSTRIDE + o]
              + Yb[(size_t)i1 * YSTRIDE + 64 + o]
              + Yb[(size_t)i2 * YSTRIDE + 128 + o]
              + bv;
        }
        tile[o * 33 + colLocal] = v;
    }
    __syncthreads();

    const int wave  = t >> 5;
    const int lane2 = t & 31;
    float* ob = out + (size_t)b * (C_OUT * NPTS) + blockIdx.x * 32 + lane2;
    #pragma unroll
    for (int oo = 0; oo < 8; ++oo) {
        const int orow = oo * 8 + wave;
        ob[(size_t)orow * NPTS] = tile[orow * 33 + lane2];  // 128B coalesced
    }
}

// ---------------------------------------------------------------------------
extern "C" void kernel_launch(void* const* d_in, const int* in_sizes, int n_in,
                              void* d_out, int out_size, void* d_ws, size_t ws_size,
                              hipStream_t stream) {
    const float*     trees   = (const float*)d_in[0];
    const long long* indexes = (const long long*)d_in[1];
    const float*     W       = (const float*)d_in[2];
    const float*     bias    = (const float*)d_in[3];
    float*           out     = (float*)d_out;

    // Workspace layout: packed W' (104 KB) then Y (256*1024*192 f32 = 192 MB).
    float* wsw = (float*)d_ws;
    float* Y   = (float*)((char*)d_ws + (size_t)WSW_DW * 4);

    wprep_kernel<<<96, 256, 0, stream>>>(W, wsw);

    dim3 g1(8, BATCH);
    gemm_kernel<<<g1, 256, 0, stream>>>(trees, wsw, Y);

    dim3 g2(32, BATCH);
    gather_kernel<<<g2, 256, 0, stream>>>(Y, indexes, bias, out);

    // Second tuple element: indexes passed through verbatim.
    hipMemcpyAsync((char*)d_out + (size_t)BATCH * C_OUT * NPTS * 4,
                   (const void*)indexes,
                   (size_t)BATCH * 3 * MPTS * 8,
                   hipMemcpyDeviceToDevice, stream);
}